// TuckerSpectralConv3d_24300924960947
// MI455X (gfx1250) — compile-verified
//
#include <hip/hip_runtime.h>

// ---------------------------------------------------------------------------
// Tucker spectral conv 3D for MI455X (gfx1250, wave32).
// Pipeline of pruned-DFT GEMM stages, all matrix math on V_WMMA_F32_16X16X4_F32.
// Bandwidth-bound (~512MB HBM traffic); intermediates live in d_ws (L2-resident).
// The two 256MB streams (input read / output write) go through the Tensor Data
// Mover (tensor_load_to_lds / tensor_store_from_lds, TENSORcnt-tracked).
// ---------------------------------------------------------------------------

typedef float v2f __attribute__((ext_vector_type(2)));
typedef float v8f __attribute__((ext_vector_type(8)));
typedef unsigned int u32x4 __attribute__((ext_vector_type(4)));
typedef int i32x8 __attribute__((ext_vector_type(8)));
typedef int i32x4 __attribute__((ext_vector_type(4)));

#define PI2_64 0.09817477042468103f  // 2*pi/64

__device__ __forceinline__ v8f wmma4(v2f a, v2f b, v8f c) {
  // D = A(16x4 f32) * B(4x16 f32) + C(16x16 f32)
  return __builtin_amdgcn_wmma_f32_16x16x4_f32(false, a, false, b, (short)0, c,
                                               false, false);
}

// --- TDM descriptors: linear tile of n fp32 (2D tensor: dim0=n, dim1=1) -----
__device__ __forceinline__ void tdm_desc(const void* gptr, unsigned lds_byte,
                                         unsigned n, u32x4& g0, i32x8& g1) {
  const unsigned long long ga = (unsigned long long)(size_t)gptr;
  g0[0] = 1u;                                  // count=1, user mode
  g0[1] = lds_byte;                            // lds_addr
  g0[2] = (unsigned)ga;                        // global_addr[31:0]
  g0[3] = ((unsigned)(ga >> 32) & 0x01FFFFFFu) // global_addr[56:32]
          | 0x80000000u;                       // type=2 ("image")
  g1[0] = 0x00020000;                          // data_size=2 (4 bytes)
  g1[1] = (int)((n & 0xFFFFu) << 16);          // tensor_dim0[15:0]
  g1[2] = (int)((n >> 16) | (1u << 16));       // tensor_dim0[31:16] | tensor_dim1=1
  g1[3] = (int)((n & 0xFFFFu) << 16);          // tile_dim0 = n
  g1[4] = 1;                                   // tile_dim1 = 1 (tile_dim2 = 0)
  g1[5] = (int)n;                              // tensor_dim0_stride[31:0]
  g1[6] = 0;
  g1[7] = 0;
}

__device__ __forceinline__ void tdm_load_linear(const void* g, unsigned lds_byte,
                                                unsigned n) {
  u32x4 g0; i32x8 g1;
  i32x4 gz4 = {0, 0, 0, 0};
  i32x8 gz8 = {0, 0, 0, 0, 0, 0, 0, 0};
  tdm_desc(g, lds_byte, n, g0, g1);
  __builtin_amdgcn_tensor_load_to_lds(g0, g1, gz4, gz4, gz8, 0);
}

__device__ __forceinline__ void tdm_store_linear(void* g, unsigned lds_byte,
                                                 unsigned n) {
  u32x4 g0; i32x8 g1;
  i32x4 gz4 = {0, 0, 0, 0};
  i32x8 gz8 = {0, 0, 0, 0, 0, 0, 0, 0};
  tdm_desc(g, lds_byte, n, g0, g1);
  __builtin_amdgcn_tensor_store_from_lds(g0, g1, gz4, gz4, gz8, 0);
}

// Workspace layout (floats).  Peak ~201 MB.
static constexpr size_t S1_PLANE = 16777216;  // B*Cin*Z*H*16   (re or im)
static constexpr size_t S2_PLANE = 4194304;   // B*Cin*Z*16*16
static constexpr size_t S3_PLANE = 2097152;   // B*32*16*16*Cin
static constexpr size_t Y_PLANE  = 2097152;   // B*Cout*16*16*32
static constexpr size_t OFF_S1RE = 0;                         // reused as S6
static constexpr size_t OFF_S1IM = OFF_S1RE + S1_PLANE;
static constexpr size_t OFF_S2RE = OFF_S1IM + S1_PLANE;       // reused as S5
static constexpr size_t OFF_S2IM = OFF_S2RE + S2_PLANE;
static constexpr size_t OFF_S3RE = OFF_S2IM + S2_PLANE;
static constexpr size_t OFF_S3IM = OFF_S3RE + S3_PLANE;
static constexpr size_t OFF_YRE  = OFF_S3IM + S3_PLANE;
static constexpr size_t OFF_YIM  = OFF_YRE + Y_PLANE;
static constexpr size_t OFF_PART = OFF_YIM + Y_PLANE;         // 1024 floats
static constexpr size_t OFF_SCAL = OFF_PART + 1024;           // 4 floats

// ---------------------------------------------------------------------------
// core sums: deterministic two-pass tree reduction (no float atomics).
// ---------------------------------------------------------------------------
__global__ void k_core_reduce1(const float* __restrict__ c0, const float* __restrict__ c1,
                               const float* __restrict__ c2, const float* __restrict__ c3,
                               float* __restrict__ partial) {
  __shared__ float sm[256];
  const int arr = blockIdx.x >> 8;   // 0..3
  const int blk = blockIdx.x & 255;
  const float* src = (arr == 0) ? c0 : (arr == 1) ? c1 : (arr == 2) ? c2 : c3;
  const int base = blk * 4096 + threadIdx.x * 16;  // 16^5 = 1,048,576 elements
  float s = 0.f;
#pragma unroll
  for (int i = 0; i < 16; ++i) s += src[base + i];
  sm[threadIdx.x] = s;
  __syncthreads();
  for (int st = 128; st > 0; st >>= 1) {
    if ((int)threadIdx.x < st) sm[threadIdx.x] += sm[threadIdx.x + st];
    __syncthreads();
  }
  if (threadIdx.x == 0) partial[arr * 256 + blk] = sm[0];
}

__global__ void k_core_reduce2(const float* __restrict__ partial, float* __restrict__ scal) {
  __shared__ float sm[256];
  sm[threadIdx.x] = partial[blockIdx.x * 256 + threadIdx.x];
  __syncthreads();
  for (int st = 128; st > 0; st >>= 1) {
    if ((int)threadIdx.x < st) sm[threadIdx.x] += sm[threadIdx.x + st];
    __syncthreads();
  }
  if (threadIdx.x == 0) scal[blockIdx.x] = sm[0];
}

// ---------------------------------------------------------------------------
// Stage 1: real forward DFT along W (64 -> 16 freqs).
// Lines l = (b,c,z,h), contiguous W.  Out: S1[(bcz*16+m)*64 + h].
// One wave per 16-line tile; tile (4KB, contiguous) DMA'd to LDS by the TDM.
// ---------------------------------------------------------------------------
__global__ void k_fwd_w(const float* __restrict__ x,
                        float* __restrict__ s1re, float* __restrict__ s1im) {
  __shared__ float lds[8][1024];
  const int wv = threadIdx.x >> 5, lane = threadIdx.x & 31;
  const int rowm = lane & 15, kh = lane >> 4, col = lane & 15;
  const long l0 = ((long)blockIdx.x * 8 + wv) * 16;

  tdm_load_linear(x + l0 * 64, (unsigned)(size_t)&lds[wv][0], 1024);
  __builtin_amdgcn_s_wait_tensorcnt(0);

  v8f ar = {0,0,0,0,0,0,0,0}, ai = {0,0,0,0,0,0,0,0};
  for (int k0 = 0; k0 < 64; k0 += 4) {
    v2f ac, as, bf;
#pragma unroll
    for (int j = 0; j < 2; ++j) {
      const int k = k0 + kh * 2 + j;
      float sn, cs;
      __sincosf(-PI2_64 * (float)((rowm * k) & 63), &sn, &cs);
      ac[j] = cs; as[j] = sn;              // e^{-i theta}
      bf[j] = lds[wv][col * 64 + k];
    }
    ar = wmma4(ac, bf, ar);
    ai = wmma4(as, bf, ai);
  }
  const long l = l0 + col;
  const long bcz = l >> 6, h = l & 63;
#pragma unroll
  for (int i = 0; i < 8; ++i) {
    const int m = kh * 8 + i;
    const long a = (bcz * 16 + m) * 64 + h;
    s1re[a] = ar[i]; s1im[a] = ai[i];
  }
}

// ---------------------------------------------------------------------------
// Stage 2: complex forward DFT along H (64 -> 16).  Lines l = (bcz, mw).
// In: S1[l*64 + h].  Out: S2[((bc*16+mh)*16+mw)*64 + z].
// ---------------------------------------------------------------------------
__global__ void k_fwd_h(const float* __restrict__ s1re, const float* __restrict__ s1im,
                        float* __restrict__ s2re, float* __restrict__ s2im) {
  const int wv = threadIdx.x >> 5, lane = threadIdx.x & 31;
  const int rowm = lane & 15, kh = lane >> 4, col = lane & 15;
  const long l0 = ((long)blockIdx.x * 8 + wv) * 16;

  v8f cre = {0,0,0,0,0,0,0,0}, cim = {0,0,0,0,0,0,0,0};
  for (int k0 = 0; k0 < 64; k0 += 4) {
    v2f ac, as, bre, bim;
#pragma unroll
    for (int j = 0; j < 2; ++j) {
      const int k = k0 + kh * 2 + j;
      float sn, cs;
      __sincosf(-PI2_64 * (float)((rowm * k) & 63), &sn, &cs);
      ac[j] = cs; as[j] = sn;
      const long ba = (l0 + col) * 64 + k;
      bre[j] = s1re[ba]; bim[j] = s1im[ba];
    }
    const v2f asn = -as;
    cre = wmma4(ac, bre, cre);  cre = wmma4(asn, bim, cre);
    cim = wmma4(ac, bim, cim);  cim = wmma4(as, bre, cim);
  }
  const long l = l0 + col;
  const long bcz = l >> 4, mw = l & 15;
  const long bc = bcz >> 6, z = bcz & 63;
#pragma unroll
  for (int i = 0; i < 8; ++i) {
    const int m = kh * 8 + i;
    const long a = ((bc * 16 + m) * 16 + mw) * 64 + z;
    s2re[a] = cre[i]; s2im[a] = cim[i];
  }
}

// ---------------------------------------------------------------------------
// Stage 3: complex forward DFT along Z (64 -> 32 freqs: 0..15 and 48..63).
// Lines l = (bc, mh, mw).  In: S2[l*64 + z].
// Out: S3[(b*8192 + zf32*256 + mh*16 + mw)*64 + c]   (c contiguous for GEMM).
// ---------------------------------------------------------------------------
__global__ void k_fwd_z(const float* __restrict__ s2re, const float* __restrict__ s2im,
                        float* __restrict__ s3re, float* __restrict__ s3im) {
  const int wv = threadIdx.x >> 5, lane = threadIdx.x & 31;
  const int rowm = lane & 15, kh = lane >> 4, col = lane & 15;
  const long gw = (long)blockIdx.x * 8 + wv;
  const long l0 = (gw >> 1) * 16;
  const int grp = (int)(gw & 1);
  const int f = (grp ? 48 : 0) + rowm;  // actual frequency for this row

  v8f cre = {0,0,0,0,0,0,0,0}, cim = {0,0,0,0,0,0,0,0};
  for (int k0 = 0; k0 < 64; k0 += 4) {
    v2f ac, as, bre, bim;
#pragma unroll
    for (int j = 0; j < 2; ++j) {
      const int k = k0 + kh * 2 + j;
      float sn, cs;
      __sincosf(-PI2_64 * (float)((f * k) & 63), &sn, &cs);
      ac[j] = cs; as[j] = sn;
      const long ba = (l0 + col) * 64 + k;
      bre[j] = s2re[ba]; bim[j] = s2im[ba];
    }
    const v2f asn = -as;
    cre = wmma4(ac, bre, cre);  cre = wmma4(asn, bim, cre);
    cim = wmma4(ac, bim, cim);  cim = wmma4(as, bre, cim);
  }
  const long l = l0 + col;
  const long bc = l >> 8, mh = (l >> 4) & 15, mw = l & 15;
  const long b = bc >> 6, c = bc & 63;
#pragma unroll
  for (int i = 0; i < 8; ++i) {
    const int zf32 = grp * 16 + kh * 8 + i;
    const long a = (b * 8192 + (long)zf32 * 256 + mh * 16 + mw) * 64 + c;
    s3re[a] = cre[i]; s3im[a] = cim[i];
  }
}

// ---------------------------------------------------------------------------
// Stage 4: spectral mixing.  One wave per (b, 16-mode tile); modes share zf,hf.
//   T = X * U_in   (16 modes x 64 ch x 16 rank, complex x real)
//   T' = s * g .* T   (per-mode separable scale, complex scalar s)
//   Y = T' * U_out^T  (16 modes x 16 rank x 64 out-ch)
// C-frag -> A-frag relayout via LDS.  Out: Y[((b*64+o)*256+hf*16+wf)*32+zf].
// ---------------------------------------------------------------------------
__global__ void k_spectral(const float* __restrict__ s3re, const float* __restrict__ s3im,
                           const float* __restrict__ Uin, const float* __restrict__ Uout,
                           const float* __restrict__ Uz, const float* __restrict__ Uh,
                           const float* __restrict__ Uw, const float* __restrict__ Uz2,
                           const float* __restrict__ Uh2, const float* __restrict__ scal,
                           float* __restrict__ yre, float* __restrict__ yim) {
  __shared__ float lt[8][512];  // per-wave: [wf*16+r] re | 256 + im
  const int wv = threadIdx.x >> 5, lane = threadIdx.x & 31;
  const int rowm = lane & 15, kh = lane >> 4, col = lane & 15;
  const long gw = (long)blockIdx.x * 8 + wv;
  const int b = (int)(gw >> 9), mt = (int)(gw & 511);
  const int mode0 = mt * 16, zf = mt >> 4, hf = mt & 15;
  const long abase = ((long)b * 8192 + mode0) * 64;

  v8f tre = {0,0,0,0,0,0,0,0}, tim = {0,0,0,0,0,0,0,0};
  for (int k0 = 0; k0 < 64; k0 += 4) {
    v2f are, aim, bf;
#pragma unroll
    for (int j = 0; j < 2; ++j) {
      const int k = k0 + kh * 2 + j;
      const long aa = abase + (long)rowm * 64 + k;
      are[j] = s3re[aa]; aim[j] = s3im[aa];
      bf[j] = Uin[k * 16 + col];
    }
    tre = wmma4(are, bf, tre);
    tim = wmma4(aim, bf, tim);
  }

  // branchless per-wave selection (zf uniform within wave)
  const bool lo = (zf < 16);
  const float sr = lo ? scal[0] : scal[2];
  const float si = lo ? scal[1] : scal[3];
  const float uzv = lo ? Uz[zf * 16 + col] : Uz2[(zf - 16) * 16 + col];
  const float* UhSel = lo ? Uh : Uh2;
  const float uwv = Uw[hf * 16 + col];  // U_w applies to the h-frequency index

#pragma unroll
  for (int i = 0; i < 8; ++i) {
    const int wf = kh * 8 + i;
    const float g = uzv * uwv * UhSel[wf * 16 + col];  // U_h applies to w-freq
    const float a_ = tre[i], b_ = tim[i];
    lt[wv][wf * 16 + col]       = g * (sr * a_ - si * b_);
    lt[wv][256 + wf * 16 + col] = g * (sr * b_ + si * a_);
  }
  __syncthreads();

  for (int og = 0; og < 4; ++og) {
    v8f yr = {0,0,0,0,0,0,0,0}, yi = {0,0,0,0,0,0,0,0};
    for (int r0 = 0; r0 < 16; r0 += 4) {
      v2f are, aim, bf;
#pragma unroll
      for (int j = 0; j < 2; ++j) {
        const int r = r0 + kh * 2 + j;
        are[j] = lt[wv][rowm * 16 + r];
        aim[j] = lt[wv][256 + rowm * 16 + r];
        bf[j] = Uout[(og * 16 + col) * 16 + r];
      }
      yr = wmma4(are, bf, yr);
      yi = wmma4(aim, bf, yi);
    }
    const int o = og * 16 + col;
    const long obase = ((long)b * 64 + o) * 256 + hf * 16;
#pragma unroll
    for (int i = 0; i < 8; ++i) {
      const int wf = kh * 8 + i;
      const long a = (obase + wf) * 32 + zf;
      yre[a] = yr[i]; yim[a] = yi[i];
    }
  }
}

// ---------------------------------------------------------------------------
// Stage 5: inverse DFT along Z (32 nonzero freqs -> 64).  Lines l=(b,o,hf,wf).
// In: Y[l*32 + zf].  Out: S5[(boz)*256 + wf*16 + hf].
// ---------------------------------------------------------------------------
__global__ void k_inv_z(const float* __restrict__ yre, const float* __restrict__ yim,
                        float* __restrict__ s5re, float* __restrict__ s5im) {
  const int wv = threadIdx.x >> 5, lane = threadIdx.x & 31;
  const int rowm = lane & 15, kh = lane >> 4, col = lane & 15;
  const long gw = (long)blockIdx.x * 8 + wv;
  const long l0 = (gw >> 2) * 16;
  const int mg = (int)(gw & 3);
  const int zc = mg * 16 + rowm;  // output z coordinate for this A-row

  v8f cre = {0,0,0,0,0,0,0,0}, cim = {0,0,0,0,0,0,0,0};
  for (int k0 = 0; k0 < 32; k0 += 4) {
    v2f ac, as, bre, bim;
#pragma unroll
    for (int j = 0; j < 2; ++j) {
      const int k = k0 + kh * 2 + j;
      const int f = k + ((k >= 16) ? 32 : 0);  // freqs 0..15, 48..63
      float sn, cs;
      __sincosf(PI2_64 * (float)((zc * f) & 63), &sn, &cs);
      ac[j] = cs; as[j] = sn;  // e^{+i theta}
      const long ba = (l0 + col) * 32 + k;
      bre[j] = yre[ba]; bim[j] = yim[ba];
    }
    const v2f asn = -as;
    cre = wmma4(ac, bre, cre);  cre = wmma4(asn, bim, cre);
    cim = wmma4(ac, bim, cim);  cim = wmma4(as, bre, cim);
  }
  const long l = l0 + col;
  const int wf = (int)(l & 15), hfv = (int)((l >> 4) & 15);
  const int o = (int)((l >> 8) & 63), b = (int)(l >> 14);
#pragma unroll
  for (int i = 0; i < 8; ++i) {
    const int z = mg * 16 + kh * 8 + i;
    const long boz = ((long)(b * 64 + o)) * 64 + z;
    const long a = boz * 256 + wf * 16 + hfv;
    s5re[a] = cre[i]; s5im[a] = cim[i];
  }
}

// ---------------------------------------------------------------------------
// Stage 6: inverse DFT along H (16 nonzero freqs -> 64).  Lines l=(boz, wf).
// In: S5[l*16 + hf].  Out: S6[(boz*64 + h)*16 + wf].
// ---------------------------------------------------------------------------
__global__ void k_inv_h(const float* __restrict__ s5re, const float* __restrict__ s5im,
                        float* __restrict__ s6re, float* __restrict__ s6im) {
  const int wv = threadIdx.x >> 5, lane = threadIdx.x & 31;
  const int rowm = lane & 15, kh = lane >> 4, col = lane & 15;
  const long gw = (long)blockIdx.x * 8 + wv;
  const long l0 = (gw >> 2) * 16;
  const int mg = (int)(gw & 3);
  const int hc = mg * 16 + rowm;

  v8f cre = {0,0,0,0,0,0,0,0}, cim = {0,0,0,0,0,0,0,0};
  for (int k0 = 0; k0 < 16; k0 += 4) {
    v2f ac, as, bre, bim;
#pragma unroll
    for (int j = 0; j < 2; ++j) {
      const int k = k0 + kh * 2 + j;
      float sn, cs;
      __sincosf(PI2_64 * (float)((hc * k) & 63), &sn, &cs);
      ac[j] = cs; as[j] = sn;
      const long ba = (l0 + col) * 16 + k;
      bre[j] = s5re[ba]; bim[j] = s5im[ba];
    }
    const v2f asn = -as;
    cre = wmma4(ac, bre, cre);  cre = wmma4(asn, bim, cre);
    cim = wmma4(ac, bim, cim);  cim = wmma4(as, bre, cim);
  }
  const long l = l0 + col;
  const int wf = (int)(l & 15);
  const long boz = l >> 4;
#pragma unroll
  for (int i = 0; i < 8; ++i) {
    const int h = mg * 16 + kh * 8 + i;
    const long a = (boz * 64 + h) * 16 + wf;
    s6re[a] = cre[i]; s6im[a] = cim[i];
  }
}

// ---------------------------------------------------------------------------
// Stage 7: real inverse DFT along W (16 nonzero rfft freqs -> 64 reals).
// out[w] = sum_wf c_wf/N^3 * (cos(th)*Re - sin(th)*Im),  c_0=1, c_wf=2.
// One wave covers all 64 w rows for 16 lines; LDS transpose, then the 4KB
// contiguous result tile is DMA'd out by the TDM (ENDPGM waits TENSORcnt).
// ---------------------------------------------------------------------------
__global__ void k_inv_w(const float* __restrict__ s6re, const float* __restrict__ s6im,
                        float* __restrict__ out) {
  __shared__ float lds[8][1024];
  const int wv = threadIdx.x >> 5, lane = threadIdx.x & 31;
  const int rowm = lane & 15, kh = lane >> 4, col = lane & 15;
  const long l0 = ((long)blockIdx.x * 8 + wv) * 16;
  const float inv = 1.f / 262144.f;  // 1 / 64^3

  for (int mg = 0; mg < 4; ++mg) {
    const int wc = mg * 16 + rowm;
    v8f acc = {0,0,0,0,0,0,0,0};
    for (int k0 = 0; k0 < 16; k0 += 4) {
      v2f a1, a2, bre, bim;
#pragma unroll
      for (int j = 0; j < 2; ++j) {
        const int k = k0 + kh * 2 + j;
        float sn, cs;
        __sincosf(PI2_64 * (float)((wc * k) & 63), &sn, &cs);
        const float scl = ((k == 0) ? 1.f : 2.f) * inv;
        a1[j] = cs * scl; a2[j] = -sn * scl;
        const long ba = (l0 + col) * 16 + k;
        bre[j] = s6re[ba]; bim[j] = s6im[ba];
      }
      acc = wmma4(a1, bre, acc);
      acc = wmma4(a2, bim, acc);
    }
#pragma unroll
    for (int i = 0; i < 8; ++i)
      lds[wv][col * 64 + mg * 16 + kh * 8 + i] = acc[i];
  }
  __syncthreads();  // flush DS writes before the TDM reads LDS
  tdm_store_linear(out + l0 * 64, (unsigned)(size_t)&lds[wv][0], 1024);
  // S_ENDPGM performs an implicit wait-idle (covers TENSORcnt).
}

// ---------------------------------------------------------------------------
extern "C" void kernel_launch(void* const* d_in, const int* in_sizes, int n_in,
                              void* d_out, int out_size, void* d_ws, size_t ws_size,
                              hipStream_t stream) {
  (void)in_sizes; (void)n_in; (void)out_size; (void)ws_size;
  const float* x     = (const float*)d_in[0];
  const float* Uin   = (const float*)d_in[1];
  const float* Uout  = (const float*)d_in[2];
  const float* Uz    = (const float*)d_in[3];
  const float* Uh    = (const float*)d_in[4];
  const float* Uw    = (const float*)d_in[5];
  const float* Uz2   = (const float*)d_in[6];
  const float* Uh2   = (const float*)d_in[7];
  const float* coreR = (const float*)d_in[8];
  const float* coreI = (const float*)d_in[9];
  const float* c2R   = (const float*)d_in[10];
  const float* c2I   = (const float*)d_in[11];
  float* ws  = (float*)d_ws;
  float* out = (float*)d_out;

  float* s1re = ws + OFF_S1RE;  float* s1im = ws + OFF_S1IM;
  float* s2re = ws + OFF_S2RE;  float* s2im = ws + OFF_S2IM;
  float* s3re = ws + OFF_S3RE;  float* s3im = ws + OFF_S3IM;
  float* yre  = ws + OFF_YRE;   float* yim  = ws + OFF_YIM;
  float* part = ws + OFF_PART;  float* scal = ws + OFF_SCAL;

  k_core_reduce1<<<1024, 256, 0, stream>>>(coreR, coreI, c2R, c2I, part);
  k_core_reduce2<<<4, 256, 0, stream>>>(part, scal);

  k_fwd_w<<<8192, 256, 0, stream>>>(x, s1re, s1im);
  k_fwd_h<<<2048, 256, 0, stream>>>(s1re, s1im, s2re, s2im);
  k_fwd_z<<<1024, 256, 0, stream>>>(s2re, s2im, s3re, s3im);
  k_spectral<<<256, 256, 0, stream>>>(s3re, s3im, Uin, Uout, Uz, Uh, Uw, Uz2,
                                      Uh2, scal, yre, yim);
  k_inv_z<<<2048, 256, 0, stream>>>(yre, yim, s2re, s2im);   // S5 reuses S2
  k_inv_h<<<8192, 256, 0, stream>>>(s2re, s2im, s1re, s1im); // S6 reuses S1
  k_inv_w<<<8192, 256, 0, stream>>>(s1re, s1im, out);
}